// GAT_74852690035322
// MI455X (gfx1250) — compile-verified
//
#include <hip/hip_runtime.h>
#include <math.h>

typedef __attribute__((ext_vector_type(2))) float v2f;
typedef __attribute__((ext_vector_type(8))) float v8f;

#define LEAKY(x) ((x) >= 0.0f ? (x) : 0.2f * (x))
#define SOFT_EPS 1e-16f

// =====================================================================
// GEMM: out[M,ncols_store] = X[M,K] @ W[K,NCOLS]   (fp32 WMMA 16x16x4)
// NCOLS is compile time -> unconditional B loads with immediate offsets,
// no exec masking in the K loop. One wave -> 16 rows x (NT*16) cols.
// A layout (16x4 f32): lanes 0-15 hold K=0/1 in v2f.{x,y}; lanes 16-31 K=2/3.
// B layout (4x16 f32): same striping; C/D: VGPR v -> row v + 8*(lane>>4).
// =====================================================================
template <int NT, int NCOLS>
__global__ __launch_bounds__(128) void gemm_wmma_f32(
    const float* __restrict__ X, const float* __restrict__ W,
    float* __restrict__ out, int M, int K, int ncols_store) {
  const int lane = threadIdx.x & 31;
  const int wave = threadIdx.x >> 5;
  const int rowTile = blockIdx.x * 4 + wave;
  if (rowTile * 16 >= M) return;  // whole wave exits uniformly (EXEC stays all-1)
  const int colBase = blockIdx.y * (NT * 16);
  const int m = lane & 15;
  const int kh = lane >> 4;  // 0 -> K pair {0,1}; 1 -> K pair {2,3}
  const int row = rowTile * 16 + m;

  v8f acc[NT];
#pragma unroll
  for (int t = 0; t < NT; ++t) acc[t] = (v8f)(0.0f);

  const float* xrow = X + (size_t)row * K;
  const float* wcol = W + colBase + m;  // per-lane B column pointer
  for (int k = 0; k < K; k += 4) {
    const int kk = k + 2 * kh;
    const v2f a = *(const v2f*)(xrow + kk);  // 8B aligned (kk even)
    const float* wr = wcol + (size_t)kk * NCOLS;
#pragma unroll
    for (int t = 0; t < NT; ++t) {
      v2f b;
      b.x = wr[t * 16];
      b.y = wr[t * 16 + NCOLS];
      acc[t] = __builtin_amdgcn_wmma_f32_16x16x4_f32(
          /*neg_a=*/false, a, /*neg_b=*/false, b,
          /*c_mod=*/(short)0, acc[t], /*reuse_a=*/false, /*reuse_b=*/false);
    }
  }

#pragma unroll
  for (int t = 0; t < NT; ++t) {
    const int col = colBase + t * 16 + m;
    if (col >= ncols_store) continue;  // runs once, outside the K loop
#pragma unroll
    for (int v = 0; v < 8; ++v) {
      const int r = rowTile * 16 + v + 8 * kh;
      out[(size_t)r * ncols_store + col] = acc[t][v];
    }
  }
}

// Zero-pad W2 [256x40] -> Wp [256x48] so the layer-2 GEMM needs no load guard.
__global__ void pad_w2(const float* __restrict__ W2, float* __restrict__ Wp) {
  const int i = blockIdx.x * blockDim.x + threadIdx.x;
  if (i >= 256 * 48) return;
  const int r = i / 48, c = i % 48;
  Wp[i] = (c < 40) ? W2[r * 40 + c] : 0.0f;
}

// =====================================================================
// a_src[n,h] = sum_c h[n,h,c]*att_src[h,c]; a_dst likewise.
// Also seeds the segment max with the self-loop score (self-loop edge
// (i,i) always exists), fusing what used to be a separate init pass.
// =====================================================================
__global__ void attn_coeff(const float* __restrict__ h,
                           const float* __restrict__ att_s,
                           const float* __restrict__ att_d,
                           float* __restrict__ a_s, float* __restrict__ a_d,
                           float* __restrict__ m, int n, int heads, int ch) {
  const int i = blockIdx.x * blockDim.x + threadIdx.x;  // node*heads + head
  if (i >= n * heads) return;
  const int head = i % heads;
  const float* hv = h + (size_t)i * ch;
  const float* as = att_s + (size_t)head * ch;
  const float* ad = att_d + (size_t)head * ch;
  float s0 = 0.0f, s1 = 0.0f;
  for (int c = 0; c < ch; ++c) {
    const float v = hv[c];
    s0 += v * as[c];
    s1 += v * ad[c];
  }
  a_s[i] = s0;
  a_d[i] = s1;
  m[i] = LEAKY(s0 + s1);  // self-loop seed for the running segment max
}

template <int H>
__global__ void edge_max(const long long* __restrict__ ei, long long E,
                         const float* __restrict__ a_s,
                         const float* __restrict__ a_d, float* __restrict__ m) {
  const long long idx = blockIdx.x * (long long)blockDim.x + threadIdx.x;
  if (idx >= E * H) return;
  const long long e = idx / H;
  const int h = (int)(idx % H);
  const long long s = ei[e], d = ei[E + e];
  const float v = LEAKY(a_s[s * H + h] + a_d[d * H + h]);
  __hip_atomic_fetch_max(&m[d * H + h], v, __ATOMIC_RELAXED,
                         __HIP_MEMORY_SCOPE_AGENT);
}

// s init: self-loop term of the softmax denominator (needs final m).
__global__ void seg_init_sum(const float* __restrict__ a_s,
                             const float* __restrict__ a_d,
                             const float* __restrict__ m,
                             float* __restrict__ ssum, int total) {
  const int i = blockIdx.x * blockDim.x + threadIdx.x;
  if (i >= total) return;
  ssum[i] = __expf(LEAKY(a_s[i] + a_d[i]) - m[i]);
}

template <int H>
__global__ void edge_sum(const long long* __restrict__ ei, long long E,
                         const float* __restrict__ a_s,
                         const float* __restrict__ a_d,
                         const float* __restrict__ m,
                         float* __restrict__ ssum) {
  const long long idx = blockIdx.x * (long long)blockDim.x + threadIdx.x;
  if (idx >= E * H) return;
  const long long e = idx / H;
  const int h = (int)(idx % H);
  const long long s = ei[e], d = ei[E + e];
  const float v = LEAKY(a_s[s * H + h] + a_d[d * H + h]);
  atomicAdd(&ssum[d * H + h], __expf(v - m[d * H + h]));
}

// out init: self-loop contribution alpha_self * h[i].
template <int HC, int C>
__global__ void agg_init(const float* __restrict__ a_s,
                         const float* __restrict__ a_d,
                         const float* __restrict__ m,
                         const float* __restrict__ ssum,
                         const float* __restrict__ hmat,
                         float* __restrict__ out, int n) {
  const size_t i = (size_t)blockIdx.x * blockDim.x + threadIdx.x;
  if (i >= (size_t)n * HC) return;
  const int node = (int)(i / HC);
  const int rem = (int)(i % HC);
  const int head = rem / C;
  const int ih = node * (HC / C) + head;
  const float v = LEAKY(a_s[ih] + a_d[ih]);
  const float alpha = __expf(v - m[ih]) / (ssum[ih] + SOFT_EPS);
  out[i] = alpha * hmat[i];
}

// Layer-1 aggregate: wave per edge; lane owns 8 contiguous channels
// (head = lane>>2, since 32 channels/head). h[src] gather is a fully
// coalesced 1KB line per wave; 8 f32 atomics/lane into L2-resident out.
__global__ __launch_bounds__(256) void edge_aggregate_l1(
    const long long* __restrict__ ei, long long E,
    const float* __restrict__ a_s, const float* __restrict__ a_d,
    const float* __restrict__ m, const float* __restrict__ ssum,
    const float* __restrict__ hmat, float* __restrict__ out) {
  const long long e = blockIdx.x * 8LL + (threadIdx.x >> 5);
  if (e >= E) return;
  const int lane = threadIdx.x & 31;
  const long long s = ei[e], d = ei[E + e];
  const int h = lane >> 2;
  const float v = LEAKY(a_s[s * 8 + h] + a_d[d * 8 + h]);
  const float alpha = __expf(v - m[d * 8 + h]) / (ssum[d * 8 + h] + SOFT_EPS);
  const float4* hp = (const float4*)(hmat + (size_t)s * 256 + lane * 8);
  const float4 v0 = hp[0], v1 = hp[1];
  float* op = out + (size_t)d * 256 + lane * 8;
  atomicAdd(op + 0, alpha * v0.x);
  atomicAdd(op + 1, alpha * v0.y);
  atomicAdd(op + 2, alpha * v0.z);
  atomicAdd(op + 3, alpha * v0.w);
  atomicAdd(op + 4, alpha * v1.x);
  atomicAdd(op + 5, alpha * v1.y);
  atomicAdd(op + 6, alpha * v1.z);
  atomicAdd(op + 7, alpha * v1.w);
}

// Layer-2 aggregate: 1 head, 40 channels; wave per edge.
__global__ __launch_bounds__(256) void edge_aggregate_l2(
    const long long* __restrict__ ei, long long E,
    const float* __restrict__ a_s, const float* __restrict__ a_d,
    const float* __restrict__ m, const float* __restrict__ ssum,
    const float* __restrict__ hmat, float* __restrict__ out) {
  const long long e = blockIdx.x * 8LL + (threadIdx.x >> 5);
  if (e >= E) return;
  const int lane = threadIdx.x & 31;
  const long long s = ei[e], d = ei[E + e];
  const float v = LEAKY(a_s[s] + a_d[d]);
  const float alpha = __expf(v - m[d]) / (ssum[d] + SOFT_EPS);
  for (int c = lane; c < 40; c += 32)
    atomicAdd(out + (size_t)d * 40 + c, alpha * hmat[(size_t)s * 40 + c]);
}

// bias + ELU (alpha=1), in place.  ncols is a power of two here (256).
__global__ void bias_elu(float* __restrict__ buf, const float* __restrict__ bias,
                         size_t total, int ncols) {
  const size_t i = (size_t)blockIdx.x * blockDim.x + threadIdx.x;
  if (i >= total) return;
  const float v = buf[i] + bias[i % ncols];
  buf[i] = v > 0.0f ? v : (__expf(v) - 1.0f);
}

// bias + log_softmax over 40 classes; wave per node, wave32 shuffle reduce.
__global__ __launch_bounds__(256) void bias_logsoftmax40(
    const float* __restrict__ in, const float* __restrict__ bias,
    float* __restrict__ out, int n) {
  const int node = blockIdx.x * 8 + (threadIdx.x >> 5);
  if (node >= n) return;
  const int lane = threadIdx.x & 31;
  const float* r = in + (size_t)node * 40;
  const float x0 = r[lane] + bias[lane];
  const float x1 = (lane < 8) ? (r[32 + lane] + bias[32 + lane]) : -INFINITY;
  float mx = fmaxf(x0, x1);
  for (int off = 16; off > 0; off >>= 1) mx = fmaxf(mx, __shfl_xor(mx, off, 32));
  float sum = __expf(x0 - mx) + ((lane < 8) ? __expf(x1 - mx) : 0.0f);
  for (int off = 16; off > 0; off >>= 1) sum += __shfl_xor(sum, off, 32);
  const float lse = mx + __logf(sum);
  float* o = out + (size_t)node * 40;
  o[lane] = x0 - lse;
  if (lane < 8) o[32 + lane] = x1 - lse;
}

// =====================================================================
extern "C" void kernel_launch(void* const* d_in, const int* in_sizes, int n_in,
                              void* d_out, int out_size, void* d_ws,
                              size_t ws_size, hipStream_t stream) {
  const float* x = (const float*)d_in[0];
  const long long* ei = (const long long*)d_in[1];
  const float* W1 = (const float*)d_in[2];
  const float* att_src1 = (const float*)d_in[3];
  const float* att_dst1 = (const float*)d_in[4];
  const float* b1 = (const float*)d_in[5];
  const float* W2 = (const float*)d_in[6];
  const float* att_src2 = (const float*)d_in[7];
  const float* att_dst2 = (const float*)d_in[8];
  const float* b2 = (const float*)d_in[9];
  float* out = (float*)d_out;

  const int N = in_sizes[0] / 256;
  const long long E = in_sizes[1] / 2;
  const int F1 = 256;  // HEADS*HID
  const int F2 = 40;   // NCLS

  // ---- workspace carve-out (everything we read is written first) ----
  char* ws = (char*)d_ws;
  auto carve = [&](size_t bytes) -> float* {
    float* p = (float*)ws;
    ws += (bytes + 255) & ~(size_t)255;
    return p;
  };
  float* h1 = carve((size_t)N * F1 * 4);  // x @ W1
  float* o1 = carve((size_t)N * F1 * 4);  // layer-1 output (post-ELU)
  float* as1 = carve((size_t)N * 8 * 4);
  float* ad1 = carve((size_t)N * 8 * 4);
  float* m1 = carve((size_t)N * 8 * 4);
  float* s1 = carve((size_t)N * 8 * 4);
  float* h2 = carve((size_t)N * F2 * 4);  // o1 @ Wp (padded W2)
  float* o2 = carve((size_t)N * F2 * 4);  // layer-2 pre-softmax
  float* as2 = carve((size_t)N * 4);
  float* ad2 = carve((size_t)N * 4);
  float* m2 = carve((size_t)N * 4);
  float* s2 = carve((size_t)N * 4);
  float* Wp = carve((size_t)256 * 48 * 4);  // zero-padded W2
  (void)ws_size;
  (void)n_in;
  (void)out_size;

  const int rowTiles = (N + 15) / 16;
  const dim3 gemmBlk(128);
  auto cdiv = [](long long a, long long b) { return (unsigned)((a + b - 1) / b); };

  // ================= layer 1 =================
  gemm_wmma_f32<4, 256><<<dim3(cdiv(rowTiles, 4), F1 / 64), gemmBlk, 0, stream>>>(
      x, W1, h1, N, 256, F1);
  attn_coeff<<<cdiv((long long)N * 8, 256), 256, 0, stream>>>(
      h1, att_src1, att_dst1, as1, ad1, m1, N, 8, 32);
  edge_max<8><<<cdiv(E * 8, 256), 256, 0, stream>>>(ei, E, as1, ad1, m1);
  seg_init_sum<<<cdiv((long long)N * 8, 256), 256, 0, stream>>>(as1, ad1, m1, s1,
                                                               N * 8);
  edge_sum<8><<<cdiv(E * 8, 256), 256, 0, stream>>>(ei, E, as1, ad1, m1, s1);
  agg_init<256, 32><<<cdiv((long long)N * 256, 256), 256, 0, stream>>>(
      as1, ad1, m1, s1, h1, o1, N);
  edge_aggregate_l1<<<cdiv(E, 8), 256, 0, stream>>>(ei, E, as1, ad1, m1, s1, h1, o1);
  bias_elu<<<cdiv((long long)N * 256, 256), 256, 0, stream>>>(
      o1, b1, (size_t)N * 256, 256);

  // ================= layer 2 =================
  pad_w2<<<cdiv(256 * 48, 256), 256, 0, stream>>>(W2, Wp);
  gemm_wmma_f32<3, 48><<<dim3(cdiv(rowTiles, 4), 1), gemmBlk, 0, stream>>>(
      o1, Wp, h2, N, 256, F2);
  attn_coeff<<<cdiv(N, 256), 256, 0, stream>>>(h2, att_src2, att_dst2, as2, ad2,
                                               m2, N, 1, 40);
  edge_max<1><<<cdiv(E, 256), 256, 0, stream>>>(ei, E, as2, ad2, m2);
  seg_init_sum<<<cdiv(N, 256), 256, 0, stream>>>(as2, ad2, m2, s2, N);
  edge_sum<1><<<cdiv(E, 256), 256, 0, stream>>>(ei, E, as2, ad2, m2, s2);
  agg_init<40, 40><<<cdiv((long long)N * 40, 256), 256, 0, stream>>>(
      as2, ad2, m2, s2, h2, o2, N);
  edge_aggregate_l2<<<cdiv(E, 8), 256, 0, stream>>>(ei, E, as2, ad2, m2, s2, h2, o2);
  bias_logsoftmax40<<<cdiv(N, 8), 256, 0, stream>>>(o2, b2, out, N);
}